// StyleSwap_39015482917451
// MI455X (gfx1250) — compile-verified
//
#include <hip/hip_runtime.h>
#include <hip/hip_bf16.h>
#include <stdint.h>

// ---------------------------------------------------------------------------
// StyleSwap on MI455X (gfx1250):
//   feat = content (*) patches  == GEMM [15376 x 2304] x [2304 x 3844]
//   -> fused argmax over 3844 patch scores (fp32 WMMA 16x16x4)
//   dec/intra reconstruction == 9-tap gather from style via argmax index map
// ---------------------------------------------------------------------------

#define PATCH     3
#define NSIDE     62            // (64-3)/1+1
#define NPATCH    (NSIDE*NSIDE) // 3844
#define NPAD      3856          // N padded to multiple of 16
#define CCH       256
#define KTOT      (CCH*PATCH*PATCH)   // 2304
#define BATCH     4
#define NROW      (BATCH*NSIDE*NSIDE) // 15376 rows of A  (961 tiles of 16)
#define MTILES    (NROW/16)           // 961
#define NTILES    ((NPATCH+15)/16)    // 241
#define LDA       (KTOT+4)            // LDS row stride (dwords): full-bank spread
#define SMEM_BYTES (16*LDA*4 + 128*4 + 128*4)

typedef __attribute__((ext_vector_type(2))) float v2f;
typedef __attribute__((ext_vector_type(8))) float v8f;

// ---------------------------------------------------------------------------
// Kernel 1: build B^T [KTOT][NPAD];  B[p][k] = style[c][py+ky][px+kx],
// k = (c,ky,kx) c-major, p = (py,px). Stored K-major so GEMM B loads are
// coalesced along N. Pad columns are zero (never win argmax: guarded).
// ---------------------------------------------------------------------------
__global__ void build_patches_kernel(const float* __restrict__ style,
                                     float* __restrict__ Bt) {
  int i = blockIdx.x * blockDim.x + threadIdx.x;
  if (i >= KTOT * NPAD) return;
  int k = i / NPAD;
  int p = i - k * NPAD;
  float v = 0.0f;
  if (p < NPATCH) {
    int c  = k / 9;
    int kr = k - c * 9;
    int ky = kr / 3;
    int kx = kr - 3 * ky;
    int py = p / NSIDE;
    int px = p - NSIDE * py;
    v = style[(c * 64 + (py + ky)) * 64 + (px + kx)];
  }
  Bt[i] = v;
}

// ---------------------------------------------------------------------------
// Kernel 2: fused GEMM + argmax.
// Grid: 961 workgroups x 256 threads (8 waves). Each WG owns 16 A-rows
// staged in LDS via GLOBAL_LOAD_ASYNC_TO_LDS_B32 (per-lane LDS dest address
// == im2col gather, no VGPR round-trip, tracked by ASYNCcnt). Each wave
// sweeps N-tiles of 16 with v_wmma_f32_16x16x4_f32, tracking per-row running
// argmax of score+enc_bias with branch-free selects.
// ---------------------------------------------------------------------------
__global__ void gemm_argmax_kernel(const float* __restrict__ content,
                                   const float* __restrict__ Bt,
                                   const float* __restrict__ enc_bias,
                                   int* __restrict__ out_idx) {
  extern __shared__ float smem[];
  float* As      = smem;                 // 16 * LDA floats
  float* red_val = smem + 16 * LDA;      // 8 waves * 16 rows
  int*   red_idx = (int*)(red_val + 128);

  const int tid  = threadIdx.x;
  const int wave = tid >> 5;
  const int lane = tid & 31;
  const int half = lane >> 4;            // 0: rows r, 1: rows r+8 of C tile
  const int l16  = lane & 15;
  const int row0 = blockIdx.x * 16;

  // ---- stage A tile: async gather global -> LDS (per-lane LDS address) ----
  // row m of A = im2col row (b,y,x); k = (c,ky,kx) c-major.
  for (int m = 0; m < 16; ++m) {
    int r   = row0 + m;
    int b   = r / (NSIDE * NSIDE);
    int rem = r - b * NSIDE * NSIDE;
    int y   = rem / NSIDE;
    int x   = rem - NSIDE * y;
    const float* src = content + (((size_t)b * CCH) * 64 + y) * 64 + x;
    float* dstrow = As + m * LDA;
    for (int k = tid; k < KTOT; k += 256) {
      int c  = k / 9;
      int kr = k - c * 9;
      int ky = kr / 3;
      int kx = kr - 3 * ky;
      uint64_t ga = (uint64_t)(uintptr_t)(src + (c * 64 + ky) * 64 + kx);
      uint32_t la = (uint32_t)(uintptr_t)(dstrow + k);  // low 32b = LDS offset
      asm volatile("global_load_async_to_lds_b32 %0, %1, off"
                   :: "v"(la), "v"(ga) : "memory");
    }
  }
  asm volatile("s_wait_asynccnt 0x0" ::: "memory");
  __syncthreads();

  float bestv[8];
  int   bestn[8];
#pragma unroll
  for (int r = 0; r < 8; ++r) { bestv[r] = -__builtin_inff(); bestn[r] = 0; }

  const float* aptr = As + l16 * LDA + half * 2;   // A frag: M=l16, K=half*2+{0,1}

  for (int nt = wave; nt < NTILES; nt += 8) {
    const int n0    = nt * 16;
    const int ncol  = n0 + l16;                    // C column owned by this lane
    const bool valid = (ncol < NPATCH);
    const float* bptr = Bt + (size_t)(half * 2) * NPAD + ncol;

    v8f acc = {};
    for (int k = 0; k < KTOT; k += 16) {
      // speculative prefetch of the B stream 64 K-rows ahead (global_prefetch_b8)
      __builtin_prefetch(bptr + (size_t)(k + 64) * NPAD, 0, 1);
#pragma unroll
      for (int kk = 0; kk < 16; kk += 4) {
        v2f a, b;
        a.x = aptr[k + kk];
        a.y = aptr[k + kk + 1];
        b.x = bptr[(size_t)(k + kk) * NPAD];
        b.y = bptr[(size_t)(k + kk + 1) * NPAD];
        acc = __builtin_amdgcn_wmma_f32_16x16x4_f32(
            /*neg_a=*/false, a, /*neg_b=*/false, b,
            /*c_mod=*/(short)0, acc, /*reuse_a=*/false, /*reuse_b=*/false);
      }
    }

    // unconditional (clamped) bias load; mask gates the update, not the load
    const int   bidx = ncol < NPATCH ? ncol : (NPATCH - 1);
    const float bias = enc_bias[bidx];
#pragma unroll
    for (int r = 0; r < 8; ++r) {
      float v = acc[r] + bias;
      // non-short-circuit bitwise logic -> v_cmp + v_cndmask, no exec branches
      bool upd = (bool)((int)valid &
                        ((int)(v > bestv[r]) |
                         ((int)(v == bestv[r]) & (int)(ncol < bestn[r]))));
      bestv[r] = upd ? v : bestv[r];
      bestn[r] = upd ? ncol : bestn[r];
    }
  }

  // ---- reduce the 16 lanes of each half (they share one M row per r) ----
#pragma unroll
  for (int r = 0; r < 8; ++r) {
    float v = bestv[r];
    int   n = bestn[r];
#pragma unroll
    for (int off = 1; off < 16; off <<= 1) {
      float v2 = __shfl_xor(v, off, 32);
      int   n2 = __shfl_xor(n, off, 32);
      bool t = (bool)((int)(v2 > v) |
                      ((int)(v2 == v) & (int)(n2 < n)));
      v = t ? v2 : v;
      n = t ? n2 : n;
    }
    if (l16 == 0) {
      int m = r + 8 * half;
      red_val[wave * 16 + m] = v;
      red_idx[wave * 16 + m] = n;
    }
  }
  __syncthreads();

  // ---- reduce across the 8 waves, write final index map ----
  if (tid < 16) {
    float v = red_val[tid];
    int   n = red_idx[tid];
#pragma unroll
    for (int w = 1; w < 8; ++w) {
      float v2 = red_val[w * 16 + tid];
      int   n2 = red_idx[w * 16 + tid];
      bool t = (bool)((int)(v2 > v) |
                      ((int)(v2 == v) & (int)(n2 < n)));
      v = t ? v2 : v;
      n = t ? n2 : n;
    }
    out_idx[row0 + tid] = n;
  }
}

// ---------------------------------------------------------------------------
// Kernel 3: dec/intra reconstruction as a gather.
// out[b,c,oy,ox] = (dec_bias[c] + sum_{9 taps} style[c][py+dy][px+dx])
//               / (intra_bias[c] + #valid taps)
// where p = idx[b][oy-dy][ox-dx].
// ---------------------------------------------------------------------------
__global__ void gather_out_kernel(const float* __restrict__ style,
                                  const int* __restrict__ idxmap,
                                  const float* __restrict__ dec_bias,
                                  const float* __restrict__ intra_bias,
                                  float* __restrict__ out) {
  int i = blockIdx.x * blockDim.x + threadIdx.x;
  if (i >= BATCH * CCH * 64 * 64) return;
  int ox = i & 63;
  int t  = i >> 6;
  int oy = t & 63;
  t >>= 6;
  int c = t & (CCH - 1);
  int b = t >> 8;

  float num = dec_bias[c];
  float den = intra_bias[c];
#pragma unroll
  for (int dy = 0; dy < 3; ++dy) {
    int y = oy - dy;
    if ((unsigned)y >= (unsigned)NSIDE) continue;
#pragma unroll
    for (int dx = 0; dx < 3; ++dx) {
      int x = ox - dx;
      if ((unsigned)x >= (unsigned)NSIDE) continue;
      int p  = idxmap[(b * NSIDE + y) * NSIDE + x];
      int py = p / NSIDE;
      int px = p - NSIDE * py;
      num += style[(c * 64 + (py + dy)) * 64 + (px + dx)];
      den += 1.0f;
    }
  }
  out[i] = num / den;
}

// ---------------------------------------------------------------------------
extern "C" void kernel_launch(void* const* d_in, const int* in_sizes, int n_in,
                              void* d_out, int out_size, void* d_ws, size_t ws_size,
                              hipStream_t stream) {
  const float* content    = (const float*)d_in[0]; // [4,256,64,64]
  const float* style      = (const float*)d_in[1]; // [256,64,64]
  const float* enc_bias   = (const float*)d_in[2]; // [3844]
  const float* dec_bias   = (const float*)d_in[3]; // [256]
  const float* intra_bias = (const float*)d_in[4]; // [256]
  float*       out        = (float*)d_out;         // [4,256,64,64]

  float* Bt     = (float*)d_ws;                                   // KTOT*NPAD f32
  int*   idxmap = (int*)((char*)d_ws + (size_t)KTOT * NPAD * 4);  // NROW ints

  {
    int n = KTOT * NPAD;
    build_patches_kernel<<<(n + 255) / 256, 256, 0, stream>>>(style, Bt);
  }
  gemm_argmax_kernel<<<MTILES, 256, SMEM_BYTES, stream>>>(content, Bt, enc_bias, idxmap);
  {
    int n = BATCH * CCH * 64 * 64;
    gather_out_kernel<<<(n + 255) / 256, 256, 0, stream>>>(style, idxmap, dec_bias,
                                                           intra_bias, out);
  }
}